// VelocityGNN_34359738368487
// MI455X (gfx1250) — compile-verified
//
#include <hip/hip_runtime.h>

// ---------------------------------------------------------------------------
// VelocityGNN for MI455X (gfx1250, wave32).
// f32 WMMA (v_wmma_f32_16x16x4_f32) for all tiled MLPs; density MLP hoisted
// from per-edge (1.6M) to per-node (100K) => 16x less matrix work; edge phase
// fuses edge-MLP + gather(dnode[src]) * edge_feat + atomic scatter to dst.
// Weights live in registers per wave (loaded once, branch-free via clamped
// loads + selects); each wave sweeps several 16-row tiles to amortize them.
// Memory-bound: ~1 GB irregular traffic -> ~45us floor at 23.3 TB/s.
// ---------------------------------------------------------------------------

typedef __attribute__((ext_vector_type(2))) float v2f;
typedef __attribute__((ext_vector_type(8))) float v8f;

#define LDS_STRIDE 36  // pad 24->36: branch-free 32-col spill, conflict-free ds

static __device__ inline v8f wmma4(v2f a, v2f b, v8f c) {
  // 8 args: (neg_a, A, neg_b, B, c_mod, C, reuse_a, reuse_b)
  return __builtin_amdgcn_wmma_f32_16x16x4_f32(false, a, false, b, (short)0, c,
                                               false, false);
}
static __device__ inline v8f splat8(float v) {
  return (v8f){v, v, v, v, v, v, v, v};
}

// Register-resident B fragments + bias for one linear layer (DIN x 24).
// B layout for v_wmma_f32_16x16x4_f32: lane holds col n = (lane&15)+16*t,
// rows k = 2*(lane>>4)+{0,1} per K-step of 4.
template <int DIN>
struct WFrag {
  static constexpr int K4 = (DIN + 3) >> 2;
  v2f w[K4][2];
  float bias[2];
};

template <int DIN>
static __device__ inline void load_wfrag(const float* __restrict__ W,
                                         const float* __restrict__ b,
                                         WFrag<DIN>& f) {
  const int lane = threadIdx.x & 31;
  const int half = lane >> 4;
  const int nloc = lane & 15;
#pragma unroll
  for (int t = 0; t < 2; ++t) {
    const int n = nloc + 16 * t;
    const bool nok = (n < 24);
    const int nc = nok ? n : 23;  // clamped address, zeroed via select
    f.bias[t] = nok ? b[nc] : 0.0f;
#pragma unroll
    for (int kt = 0; kt < WFrag<DIN>::K4; ++kt) {
      const int k0 = kt * 4 + half * 2;
      const int k0c = (k0 < DIN) ? k0 : (DIN - 1);
      const int k1c = (k0 + 1 < DIN) ? (k0 + 1) : (DIN - 1);
      const float w0 = W[k0c * 24 + nc];  // unconditional loads, no branches
      const float w1 = W[k1c * 24 + nc];
      f.w[kt][t].x = (nok && k0 < DIN) ? w0 : 0.0f;
      f.w[kt][t].y = (nok && k0 + 1 < DIN) ? w1 : 0.0f;
    }
  }
}

// acc(16x24) = A(16xDIN) * Wfrag + bias.  A row m = lane&15; lanes<16 feed
// K={0,1}, lanes>=16 feed K={2,3} per 4-wide K step.  Clamped loads, selects.
template <int DIN, bool MASK_ROWS>
static __device__ inline void mlp_stage(const float* A, long arow0, int astride,
                                        int rows_valid, const WFrag<DIN>& f,
                                        v8f acc[2]) {
  const int lane = threadIdx.x & 31;
  const int half = lane >> 4;
  const int m = lane & 15;
  acc[0] = splat8(f.bias[0]);
  acc[1] = splat8(f.bias[1]);
  const bool mok = !MASK_ROWS || (m < rows_valid);
  const long mrow = (arow0 + (mok ? m : 0)) * (long)astride;
#pragma unroll
  for (int kt = 0; kt < WFrag<DIN>::K4; ++kt) {
    const int k0 = kt * 4 + half * 2;
    const int k0c = (k0 < DIN) ? k0 : 0;
    const int k1c = (k0 + 1 < DIN) ? (k0 + 1) : 0;
    const float a0 = A[mrow + k0c];
    const float a1 = A[mrow + k1c];
    v2f a;
    a.x = (mok && k0 < DIN) ? a0 : 0.0f;
    a.y = (mok && k0 + 1 < DIN) ? a1 : 0.0f;
    acc[0] = wmma4(a, f.w[kt][0], acc[0]);
    acc[1] = wmma4(a, f.w[kt][1], acc[1]);
  }
}

// Spill a 16x24 C/D tile (rows-across-VGPRs) to LDS row-major, ReLU'd.
// Full 32 columns written (24..31 are padding, never read back).
static __device__ inline void store_tile_lds(float* sh, const v8f acc[2]) {
  const int lane = threadIdx.x & 31;
  const int half = lane >> 4;
  const int nloc = lane & 15;
#pragma unroll
  for (int t = 0; t < 2; ++t) {
    const int n = nloc + 16 * t;
#pragma unroll
    for (int g = 0; g < 8; ++g) {
      const int m = g + 8 * half;
      sh[m * LDS_STRIDE + n] = fmaxf(acc[t][g], 0.0f);
    }
  }
}

// --------------------------- kernels ---------------------------------------

__global__ void degree_kernel(const int* __restrict__ dst, long n_edges,
                              float* __restrict__ cnt) {
  long i = (long)blockIdx.x * blockDim.x + threadIdx.x;
  if (i < n_edges) atomicAdd(&cnt[dst[i]], 1.0f);
}

// Per-node density MLP3. One wave per block; TILES x 16 nodes per wave with
// all weight fragments held in registers across tiles.
template <int DIN, int TILES>
__global__ void __launch_bounds__(32)
node_mlp_kernel(const float* __restrict__ hin, long n_rows,
                const float* __restrict__ W0, const float* __restrict__ b0,
                const float* __restrict__ W1, const float* __restrict__ b1,
                const float* __restrict__ W2, const float* __restrict__ b2,
                float* __restrict__ out) {
  __shared__ float sh[16 * LDS_STRIDE];
  WFrag<DIN> f0;
  WFrag<24> f1, f2;
  load_wfrag<DIN>(W0, b0, f0);
  load_wfrag<24>(W1, b1, f1);
  load_wfrag<24>(W2, b2, f2);

  const int lane = threadIdx.x & 31;
  const int half = lane >> 4;
  const int nloc = lane & 15;

  for (int tile = 0; tile < TILES; ++tile) {
    const long base = ((long)blockIdx.x * TILES + tile) * 16;
    if (base >= n_rows) break;  // uniform
    const int rows = (n_rows - base) >= 16 ? 16 : (int)(n_rows - base);

    v8f acc[2];
    mlp_stage<DIN, true>(hin, base, DIN, rows, f0, acc);
    store_tile_lds(sh, acc);
    __syncthreads();  // single-wave block: S_NOP + compiler fence
    mlp_stage<24, false>(sh, 0, LDS_STRIDE, 16, f1, acc);
    store_tile_lds(sh, acc);
    __syncthreads();
    mlp_stage<24, false>(sh, 0, LDS_STRIDE, 16, f2, acc);

#pragma unroll
    for (int t = 0; t < 2; ++t) {
      const int n = nloc + 16 * t;
      if (n < 24) {
#pragma unroll
        for (int g = 0; g < 8; ++g) {
          const int m = g + 8 * half;
          if (m < rows) out[(base + m) * 24 + n] = acc[t][g];
        }
      }
    }
    __syncthreads();
  }
}

// Per-edge: edge_feat = MLP3(edge_attr) via WMMA, fused with
// atomicAdd(accum[dst], edge_feat * dnode[src]).  TILES x 16 edges per wave.
template <int TILES>
__global__ void __launch_bounds__(32)
edge_msg_kernel(const float* __restrict__ eattr, const int* __restrict__ src,
                const int* __restrict__ dst, long n_edges,
                const float* __restrict__ W0, const float* __restrict__ b0,
                const float* __restrict__ W1, const float* __restrict__ b1,
                const float* __restrict__ W2, const float* __restrict__ b2,
                const float* __restrict__ dnode, float* __restrict__ accum) {
  __shared__ float sh[16 * LDS_STRIDE];
  __shared__ int ssrc[16];
  __shared__ int sdst[16];
  WFrag<3> f0;
  WFrag<24> f1, f2;
  load_wfrag<3>(W0, b0, f0);
  load_wfrag<24>(W1, b1, f1);
  load_wfrag<24>(W2, b2, f2);

  const int lane = threadIdx.x & 31;
  const int half = lane >> 4;
  const int nloc = lane & 15;

  for (int tile = 0; tile < TILES; ++tile) {
    const long base = ((long)blockIdx.x * TILES + tile) * 16;
    if (base >= n_edges) break;  // uniform
    const int rows = (n_edges - base) >= 16 ? 16 : (int)(n_edges - base);

    // prefetch next tile's streams while this tile computes
    const long nb = base + 16;
    if (tile + 1 < TILES && nb < n_edges) {  // uniform
      __builtin_prefetch(&eattr[nb * 3], 0, 1);
      __builtin_prefetch(&src[nb], 0, 1);
      __builtin_prefetch(&dst[nb], 0, 1);
    }

    if (lane < 16) {
      const bool ok = lane < rows;
      ssrc[lane] = ok ? src[base + lane] : 0;
      sdst[lane] = ok ? dst[base + lane] : -1;
    }
    __syncthreads();

    v8f acc[2];
    mlp_stage<3, true>(eattr, base, 3, rows, f0, acc);
    store_tile_lds(sh, acc);
    __syncthreads();
    mlp_stage<24, false>(sh, 0, LDS_STRIDE, 16, f1, acc);
    store_tile_lds(sh, acc);
    __syncthreads();
    mlp_stage<24, false>(sh, 0, LDS_STRIDE, 16, f2, acc);

    // gather * scatter straight from the C fragments
#pragma unroll
    for (int t = 0; t < 2; ++t) {
      const int n = nloc + 16 * t;
      if (n < 24) {
#pragma unroll
        for (int g = 0; g < 8; ++g) {
          const int m = g + 8 * half;
          const int d = sdst[m];
          if (d >= 0) {
            const float v = acc[t][g] * dnode[(long)ssrc[m] * 24 + n];
            atomicAdd(&accum[(long)d * 24 + n], v);
          }
        }
      }
    }
    __syncthreads();
  }
}

// h = accum / max(count,1), float4-vectorized (24 % 4 == 0: a 4-vector never
// straddles a node row; node id = i/6).
__global__ void normalize_kernel(const float4* __restrict__ acc,
                                 const float* __restrict__ cnt, int n_vec,
                                 float4* __restrict__ hout) {
  int i = blockIdx.x * blockDim.x + threadIdx.x;
  if (i < n_vec) {
    float4 v = acc[i];
    const float d = fmaxf(cnt[i / 6], 1.0f);
    v.x = v.x / d;
    v.y = v.y / d;
    v.z = v.z / d;
    v.w = v.w / d;
    hout[i] = v;
  }
}

// Readout: 24 -> 24 (relu) -> 3. One thread per node; tiny, HBM-bound.
__global__ void readout_kernel(const float* __restrict__ h,
                               const float* __restrict__ W0,
                               const float* __restrict__ b0,
                               const float* __restrict__ W1,
                               const float* __restrict__ b1, int n_nodes,
                               float* __restrict__ out) {
  int i = blockIdx.x * blockDim.x + threadIdx.x;
  if (i >= n_nodes) return;
  float hin[24];
#pragma unroll
  for (int j = 0; j < 24; ++j) hin[j] = h[(long)i * 24 + j];
  float mid[24];
#pragma unroll
  for (int j = 0; j < 24; ++j) {
    float s = b0[j];
#pragma unroll
    for (int k = 0; k < 24; ++k) s += hin[k] * W0[k * 24 + j];
    mid[j] = fmaxf(s, 0.0f);
  }
#pragma unroll
  for (int j = 0; j < 3; ++j) {
    float s = b1[j];
#pragma unroll
    for (int k = 0; k < 24; ++k) s += mid[k] * W1[k * 3 + j];
    out[(long)i * 3 + j] = s;
  }
}

// --------------------------- launcher --------------------------------------

static inline size_t align256(size_t x) { return (x + 255) & ~(size_t)255; }

extern "C" void kernel_launch(void* const* d_in, const int* in_sizes, int n_in,
                              void* d_out, int out_size, void* d_ws,
                              size_t ws_size, hipStream_t stream) {
  const float* x = (const float*)d_in[0];
  const int* eidx = (const int*)d_in[1];
  const float* eattr = (const float*)d_in[2];

  const int N = in_sizes[0] / 3;         // 100000 nodes
  const long E = (long)in_sizes[2] / 3;  // 1600000 edges
  const int* src = eidx;                 // edge_index[0]
  const int* dst = eidx + E;             // edge_index[1]

  // Param leaves in pytree order: per layer {density W0,b0,W1,b1,W2,b2,
  // edge W0,b0,W1,b1,W2,b2}, then readout {W0,b0,W1,b1}.
  const float* P[40];
  for (int i = 3; i < n_in && i < 43; ++i) P[i - 3] = (const float*)d_in[i];

  // Workspace carve-out
  char* ws = (char*)d_ws;
  size_t off = 0;
  float* cnt = (float*)(ws + off);
  off += align256((size_t)N * 4);
  float* hA = (float*)(ws + off);
  off += align256((size_t)N * 24 * 4);
  float* hB = (float*)(ws + off);
  off += align256((size_t)N * 24 * 4);
  float* dnode = (float*)(ws + off);
  off += align256((size_t)N * 24 * 4);

  // In-degree counts (mean aggregation denominator)
  hipMemsetAsync(cnt, 0, (size_t)N * 4, stream);
  degree_kernel<<<(int)((E + 255) / 256), 256, 0, stream>>>(dst, E, cnt);

  constexpr int NT = 2;  // node tiles per wave -> 3125 waves
  constexpr int ET = 8;  // edge tiles per wave -> 12500 waves

  const float* h = x;
  for (int l = 0; l < 3; ++l) {
    const float* const* D = &P[12 * l];       // density MLP
    const float* const* Ew = &P[12 * l + 6];  // edge MLP

    const int nblk = (N + 16 * NT - 1) / (16 * NT);
    if (l == 0)
      node_mlp_kernel<3, NT><<<nblk, 32, 0, stream>>>(
          h, N, D[0], D[1], D[2], D[3], D[4], D[5], dnode);
    else
      node_mlp_kernel<24, NT><<<nblk, 32, 0, stream>>>(
          h, N, D[0], D[1], D[2], D[3], D[4], D[5], dnode);

    hipMemsetAsync(hB, 0, (size_t)N * 24 * 4, stream);
    const int eblk = (int)((E + 16 * ET - 1) / (16 * ET));
    edge_msg_kernel<ET><<<eblk, 32, 0, stream>>>(eattr, src, dst, E, Ew[0],
                                                 Ew[1], Ew[2], Ew[3], Ew[4],
                                                 Ew[5], dnode, hB);
    normalize_kernel<<<(N * 6 + 255) / 256, 256, 0, stream>>>(
        (const float4*)hB, cnt, N * 6, (float4*)hA);
    h = hA;
  }

  readout_kernel<<<(N + 255) / 256, 256, 0, stream>>>(
      hA, P[36], P[37], P[38], P[39], N, (float*)d_out);
}